// SPDRectifiedLinear_56616258896175
// MI455X (gfx1250) — compile-verified
//
#include <hip/hip_runtime.h>

typedef float v2f __attribute__((ext_vector_type(2)));
typedef float v8f __attribute__((ext_vector_type(8)));

#define NDIM   64
#define LDA    65   // f64 row stride (pad to stagger banks)
#define LDF    68   // f32 row stride (pad: column walks hit distinct banks)
#define SWEEPS 12
#define NPAIR  32

// round-robin tournament pairing: player 63 fixed, 0..62 rotate.
__device__ __forceinline__ void pair_pq(int round, int k, int* p, int* q) {
    if (k == 0) {
        *p = 63;
        *q = round;            // round in [0,63)
    } else {
        int a = round + k;       if (a >= 63) a -= 63;
        int b = round + 63 - k;  if (b >= 63) b -= 63;
        *p = a;
        *q = b;
    }
}

__global__ __launch_bounds__(256)
void spd_eigh_recon_kernel(const float* __restrict__ x, float* __restrict__ out)
{
    // f64 working matrix (eigenvalue fidelity near the 1e-3 clamp floor)
    __shared__ double Ash[NDIM * LDA];          // 33280 B
    // f32 eigenvector accumulator
    __shared__ float  Vf[NDIM * LDF];           // 17408 B
    __shared__ double crot[NPAIR], srot[NPAIR]; //   512 B
    __shared__ float  crf[NPAIR],  srf[NPAIR];  //   256 B
    __shared__ float  wf[NDIM];                 //   256 B

    const int tid = threadIdx.x;
    const int mat = blockIdx.x;
    const float* __restrict__ xm = x   + (size_t)mat * (NDIM * NDIM);
    float*       __restrict__ om = out + (size_t)mat * (NDIM * NDIM);

    // ---- load A (f32 -> f64), V = I (f32) ----
    for (int i = tid; i < NDIM * NDIM; i += 256) {
        int r = i >> 6, c = i & 63;
        Ash[r * LDA + c] = (double)xm[i];
        Vf[r * LDF + c]  = (r == c) ? 1.0f : 0.0f;
    }
    __syncthreads();

    // ---- parallel-ordering two-sided Jacobi ----
    for (int sweep = 0; sweep < SWEEPS; ++sweep) {
        for (int round = 0; round < 63; ++round) {
            // phase 1: one thread per pair computes the rotation
            if (tid < NPAIR) {
                int p, q; pair_pq(round, tid, &p, &q);
                double app = Ash[p * LDA + p];
                double aqq = Ash[q * LDA + q];
                double apq = Ash[p * LDA + q];
                double c = 1.0, s = 0.0;
                if (fabs(apq) > 1e-300) {
                    double tau = (aqq - app) / (2.0 * apq);
                    double t   = ((tau >= 0.0) ? 1.0 : -1.0) /
                                 (fabs(tau) + sqrt(1.0 + tau * tau));
                    c = 1.0 / sqrt(1.0 + t * t);
                    s = t * c;
                }
                crot[tid] = c;        srot[tid] = s;
                crf[tid]  = (float)c; srf[tid]  = (float)s;
            }
            __syncthreads();

            // phase 2: row rotations (A <- J^T A), 32 pairs x 64 cols
            for (int idx = tid; idx < NPAIR * NDIM; idx += 256) {
                int k = idx >> 6, col = idx & 63;
                int p, q; pair_pq(round, k, &p, &q);
                double c = crot[k], s = srot[k];
                double ap = Ash[p * LDA + col];
                double aq = Ash[q * LDA + col];
                Ash[p * LDA + col] = c * ap - s * aq;
                Ash[q * LDA + col] = s * ap + c * aq;
            }
            __syncthreads();

            // phase 3: column rotations of A (A <- A J) and V (V <- V J)
            for (int idx = tid; idx < NPAIR * NDIM; idx += 256) {
                int k = idx >> 6, row = idx & 63;
                int p, q; pair_pq(round, k, &p, &q);
                double c = crot[k], s = srot[k];
                double ap = Ash[row * LDA + p];
                double aq = Ash[row * LDA + q];
                Ash[row * LDA + p] = c * ap - s * aq;
                Ash[row * LDA + q] = s * ap + c * aq;
                float cf = crf[k], sf = srf[k];
                float vp = Vf[row * LDF + p];
                float vq = Vf[row * LDF + q];
                Vf[row * LDF + p] = cf * vp - sf * vq;
                Vf[row * LDF + q] = sf * vp + cf * vq;
            }
            __syncthreads();
        }
    }

    // ---- eigenvalue clamp (f64), cast to f32 ----
    if (tid < NDIM) {
        double w = Ash[tid * LDA + tid];
        w = fmin(fmax(w, 1.0e-3), 1.0e3);
        wf[tid] = (float)w;
    }
    __syncthreads();

    // ---- out = V diag(w) V^T via V_WMMA_F32_16X16X4_F32 ----
    // A-tile: rows i0..i0+15 of (V * diag(w));  B-tile: (V^T) rows k, cols j0..j0+15.
    // 32-bit 16x4 A layout: lanes 0-15 hold M, VGPR0=K0/VGPR1=K1; lanes 16-31 hold K2/K3.
    const int wave  = tid >> 5;
    const int lane  = tid & 31;
    const int lrow  = lane & 15;
    const int khalf = (lane >> 4) << 1;   // 0 or 2

    for (int t = wave; t < 16; t += 8) {  // wave-uniform: EXEC stays all-ones
        const int i0 = (t >> 2) << 4;
        const int j0 = (t & 3) << 4;
        v8f acc = {};
        #pragma unroll
        for (int kk = 0; kk < NDIM; kk += 4) {
            const int k0 = kk + khalf;
            v2f a, b;
            a[0] = Vf[(i0 + lrow) * LDF + k0]     * wf[k0];
            a[1] = Vf[(i0 + lrow) * LDF + k0 + 1] * wf[k0 + 1];
            b[0] = Vf[(j0 + lrow) * LDF + k0];
            b[1] = Vf[(j0 + lrow) * LDF + k0 + 1];
            acc = __builtin_amdgcn_wmma_f32_16x16x4_f32(
                false, a, false, b, (short)0, acc, false, false);
        }
        // D layout: VGPR r -> M = r (lanes 0-15) / M = r+8 (lanes 16-31); N = lane&15
        const int rbase = i0 + ((lane >> 4) << 3);
        const int col   = j0 + lrow;
        #pragma unroll
        for (int r = 0; r < 8; ++r) {
            om[(rbase + r) * NDIM + col] = acc[r];
        }
    }
}

extern "C" void kernel_launch(void* const* d_in, const int* in_sizes, int n_in,
                              void* d_out, int out_size, void* d_ws, size_t ws_size,
                              hipStream_t stream) {
    const float* x = (const float*)d_in[0];
    float* out = (float*)d_out;
    const int nmat = in_sizes[0] / (NDIM * NDIM);   // 8192
    spd_eigh_recon_kernel<<<dim3(nmat), dim3(256), 0, stream>>>(x, out);
}